// Toeplizt_42769284333605
// MI455X (gfx1250) — compile-verified
//
#include <hip/hip_runtime.h>
#include <hip/hip_bf16.h>
#include <stdint.h>

// CDNA5 wave32 WMMA types
typedef __attribute__((ext_vector_type(2))) float v2f;
typedef __attribute__((ext_vector_type(8))) float v8f;

#define N_SEQ 8192
#define E_DIM 768
#define M_TILE 128
#define KFLAT_LEN (2 * N_SEQ)          // 16384 taps (index d+8191, d in [-8191, 8192])
#define KBAND (N_SEQ + M_TILE)         // 8320 floats of the diagonal band staged in LDS

// ---------------------------------------------------------------------------
// Kernel 1: flatten the Toeplitz taps into kflat[m] = T(m - 8191)
//   T(0)=zero[0], T(k>0)=pos[k-1], T(-k)=neg[(n-1)-k]
// ---------------------------------------------------------------------------
__global__ void build_kflat_kernel(const float* __restrict__ pos,
                                   const float* __restrict__ zero,
                                   const float* __restrict__ neg,
                                   float* __restrict__ kflat) {
  int m = blockIdx.x * blockDim.x + threadIdx.x;
  if (m >= KFLAT_LEN) return;
  int d = m - (N_SEQ - 1);
  float v;
  if (d == 0)            v = zero[0];
  else if (d > 0)        v = (d <= N_SEQ - 1) ? pos[d - 1] : 0.0f;  // d==8192 unused
  else                   v = neg[(N_SEQ - 1) + d];
  kflat[m] = v;
}

// ---------------------------------------------------------------------------
// Kernel 2: Out_b[i, c] = sum_j T(i-j) * X_b[j, c], computed as
//   out[i] = sum_K kflat[i + K] * x[n-1-K]     (K = n-1-j, ascending A index)
// so the WMMA A-operand pairs {K, K+1} are ascending in LDS (no reg swaps)
// and all 8 M-tiles share one DS base VGPR via immediate offsets.
// Grid: (N_SEQ/128, E_DIM/128, batch). Block: 256 threads = 8 wave32s.
// Wave w computes a 128(M) x 16(N) strip: 8 C-tiles of 16x16.
// A-operand for the whole WG/K-loop = kflat[i0 .. i0+8319] staged once in LDS
// via async global->LDS B128 (ASYNCcnt), then read as 16x4 fp32 fragments.
// ---------------------------------------------------------------------------
__launch_bounds__(256)
__global__ void toeplitz_wmma_kernel(const float* __restrict__ x,
                                     const float* __restrict__ kflat,
                                     float* __restrict__ out) {
  __shared__ float lds_k[KBAND];

  const int lane  = threadIdx.x & 31;
  const int wave  = threadIdx.x >> 5;            // 0..7
  const int i0    = blockIdx.x * M_TILE;         // M-tile base row
  const int mrow  = lane & 15;                   // M within tile (A) / N within tile (B,C)
  const int khalf = lane >> 4;                   // selects K pair (A,B)
  const int ncol  = blockIdx.y * 128 + wave * 16 + mrow;     // output column
  const size_t boff = (size_t)blockIdx.z * (size_t)N_SEQ * E_DIM;

  // ---- Stage diagonal band kflat[i0 .. i0+KBAND) into LDS (async, B128) ----
  {
    const float* gsrc = kflat + i0;                     // 512B aligned (i0 % 128 == 0)
    const uint32_t lbase = (uint32_t)(uintptr_t)(&lds_k[0]);
    for (int c = (int)threadIdx.x; c < (KBAND / 4); c += 256) {
      uint32_t laddr = lbase + (uint32_t)c * 16u;
      uint32_t goff  = (uint32_t)c * 16u;
      asm volatile("global_load_async_to_lds_b128 %0, %1, %2"
                   :
                   : "v"(laddr), "v"(goff), "s"(gsrc)
                   : "memory");
    }
    asm volatile("s_wait_asynccnt 0" ::: "memory");
    __syncthreads();
  }

  // ---- Accumulators: 8 stacked 16x16 C tiles (covers M = i0 .. i0+127) ----
  v8f acc[8];
#pragma unroll
  for (int t = 0; t < 8; ++t)
#pragma unroll
    for (int v = 0; v < 8; ++v) acc[t][v] = 0.0f;

  // A fragment LDS index (floats), band-relative, ASCENDING in K:
  //   A[M=16t+mrow, K=k0+2*khalf+v] = lds_k[16t + mrow + 2*khalf + k0 + v]
  // -> single running base per lane; 16t and v fold into DS immediate offsets.
  int aidx = mrow + 2 * khalf;                   // += 4 per k-step

  // B fragment: B[K, N] = x[n-1 - (k0 + 2*khalf + v), ncol]  (rows descend)
  const float* xptr = x + boff + (size_t)(N_SEQ - 1 - 2 * khalf) * E_DIM + ncol;

#pragma unroll 4
  for (int k0 = 0; k0 < N_SEQ; k0 += 4) {
    v2f bfrag;
    bfrag.x = xptr[0];                  // row n-1 - (k0 + 2*khalf)
    bfrag.y = xptr[-E_DIM];             // row n-1 - (k0 + 2*khalf + 1)
    xptr -= 4 * E_DIM;

#pragma unroll
    for (int t = 0; t < 8; ++t) {
      v2f afrag;
      afrag.x = lds_k[aidx + 16 * t];         // K pair lands in WMMA register order
      afrag.y = lds_k[aidx + 16 * t + 1];     // -> ds_load_2addr, no v_mov swaps
      acc[t] = __builtin_amdgcn_wmma_f32_16x16x4_f32(
          /*neg_a=*/false, afrag, /*neg_b=*/false, bfrag,
          /*c_mod=*/(short)0, acc[t], /*reuse_a=*/false, /*reuse_b=*/false);
    }
    aidx += 4;
  }

  // ---- Epilogue: C layout -> lanes 0-15: M=v, lanes 16-31: M=v+8, N=lane&15
  float* optr = out + boff + (size_t)(i0 + 8 * khalf) * E_DIM + ncol;
#pragma unroll
  for (int t = 0; t < 8; ++t) {
#pragma unroll
    for (int v = 0; v < 8; ++v) {
      optr[(size_t)(16 * t + v) * E_DIM] = acc[t][v];
    }
  }
}

// ---------------------------------------------------------------------------
extern "C" void kernel_launch(void* const* d_in, const int* in_sizes, int n_in,
                              void* d_out, int out_size, void* d_ws, size_t ws_size,
                              hipStream_t stream) {
  const float* x    = (const float*)d_in[0];   // [b, 8192, 768] f32
  const float* pos  = (const float*)d_in[1];   // [8191] f32
  const float* zero = (const float*)d_in[2];   // [1] f32
  const float* neg  = (const float*)d_in[3];   // [8191] f32
  float* out   = (float*)d_out;                // [b, 8192, 768] f32
  float* kflat = (float*)d_ws;                 // 16384 f32 = 64 KB scratch

  const int batches = in_sizes[0] / (N_SEQ * E_DIM);   // = 4

  build_kflat_kernel<<<KFLAT_LEN / 256, 256, 0, stream>>>(pos, zero, neg, kflat);

  dim3 grid(N_SEQ / M_TILE, E_DIM / 128, batches);
  toeplitz_wmma_kernel<<<grid, 256, 0, stream>>>(x, kflat, out);
}